// GraphSAGE_10239202034211
// MI455X (gfx1250) — compile-verified
//
#include <hip/hip_runtime.h>
#include <math.h>

#define N_NODES 50000
#define DEG     32
#define WIDTH   128
#define DEPTH   2
#define BATCH   4096
#define EPS     1e-12f

#define LDS_STRIDE 132   // 128 + 4 pad: row base advances 4 banks/row -> conflict-free frag reads

typedef __attribute__((ext_vector_type(2))) float v2f;
typedef __attribute__((ext_vector_type(8))) float v8f;

// ---------------------------------------------------------------- init / scatter (float4-wide)
__global__ __launch_bounds__(256) void k_zero4(float4* __restrict__ p, int n4) {
  int t = blockIdx.x * 256 + threadIdx.x;
  if (t < n4) p[t] = make_float4(0.f, 0.f, 0.f, 0.f);
}

__global__ __launch_bounds__(256) void k_scatter(const int* __restrict__ node_idx,
                                                 const float* __restrict__ feats,
                                                 float* __restrict__ H) {
  int t = blockIdx.x * 256 + threadIdx.x;            // BATCH*32 threads (float4 units)
  int i = t >> 5;                                    // batch row
  int j = t & 31;                                    // float4 column
  float4 v = ((const float4*)feats)[(size_t)i * 32 + j];
  ((float4*)H)[(size_t)node_idx[i] * 32 + j] = v;
}

// Stage a 16x128 f32 tile (rows starting at r0) from global into padded LDS.
// 256 threads, coalesced b128 loads; 2048 floats total.
__device__ __forceinline__ void stage_tile(const float* __restrict__ src,
                                           float (*dst)[LDS_STRIDE], int r0) {
  const float4* s4 = (const float4*)(src + (size_t)r0 * WIDTH);
#pragma unroll
  for (int i = 0; i < 2; ++i) {
    int idx4 = threadIdx.x + i * 256;                // 0..511 float4 index
    float4 v = s4[idx4];
    int row = idx4 >> 5;                             // (idx4*4)/128
    int col = (idx4 & 31) * 4;
    dst[row][col + 0] = v.x; dst[row][col + 1] = v.y;
    dst[row][col + 2] = v.z; dst[row][col + 3] = v.w;
  }
}

// ---------------------------------------------------------------- GEMM1: M = relu(H @ W_agg^T + b)
// One wave -> one 16x16 output tile; block = 8 waves = 16 rows x 128 cols.
// A tile staged once in LDS (kills 8x redundant global traffic); B (weights) stays L2-resident.
__global__ __launch_bounds__(256)
void k_gemm_agg(const float* __restrict__ H,      // [N][128]
                const float* __restrict__ W,      // [128][128] row-major (out x K)
                const float* __restrict__ bias,   // [128]
                float* __restrict__ out) {        // [N][128]
  __shared__ float As[16][LDS_STRIDE];

  const int lane = threadIdx.x & 31;
  const int wave = threadIdx.x >> 5;               // column tile 0..7
  const int r0   = blockIdx.x * 16;
  const int c0   = wave * 16;
  const int mn   = lane & 15;
  const int kh   = (lane >> 4) * 2;                // K sub-offset 0 or 2

  stage_tile(H, As, r0);
  __syncthreads();

  const float* brow = W + (size_t)(c0 + mn) * WIDTH; // B: one W row (=out col) per lane

  v8f acc = {};
#pragma unroll 8
  for (int k0 = 0; k0 < WIDTH; k0 += 4) {
    v2f a = *(const v2f*)&As[mn][k0 + kh];          // ds_load_b64, bank-conflict-free
    v2f b; b.x = brow[k0 + kh + 0]; b.y = brow[k0 + kh + 1];
    acc = __builtin_amdgcn_wmma_f32_16x16x4_f32(false, a, false, b,
                                                (short)0, acc, false, false);
  }

  const int col = c0 + mn;
  const float bb = bias[col];
#pragma unroll
  for (int v = 0; v < 8; ++v) {
    int row = r0 + v + 8 * (lane >> 4);
    float val = acc[v] + bb;
    out[(size_t)row * WIDTH + col] = val > 0.0f ? val : 0.0f;
  }
}

// ---------------------------------------------------------------- neighbor max-pool
// agg[n][j] = max_d M[nbd[n][d]][j]. One wave per node, float4 lanes:
// one global_load_b128 per lane pulls a full 512B row per wave. M is L2-resident.
__global__ __launch_bounds__(256)
void k_pool(const float* __restrict__ M, const int* __restrict__ nbd,
            float* __restrict__ AGG) {
  const int wave = threadIdx.x >> 5;
  const int lane = threadIdx.x & 31;
  const int n = blockIdx.x * 8 + wave;             // 6250 blocks x 8 nodes
  const int* nb = nbd + (size_t)n * DEG;           // wave-uniform -> scalar loads

  float4 m = make_float4(-__builtin_inff(), -__builtin_inff(),
                         -__builtin_inff(), -__builtin_inff());
#pragma unroll 4
  for (int d = 0; d < DEG; ++d) {
    int idx = nb[d];
    float4 v = ((const float4*)M)[(size_t)idx * 32 + lane];
    m.x = fmaxf(m.x, v.x); m.y = fmaxf(m.y, v.y);
    m.z = fmaxf(m.z, v.z); m.w = fmaxf(m.w, v.w);
  }
  ((float4*)AGG)[(size_t)n * 32 + lane] = m;
}

// ---------------------------------------------------------------- GEMM2 + L2-normalize
// h = relu([H | AGG] @ W_lin^T + b),  H_next = h / max(||h||, eps)
__global__ __launch_bounds__(256)
void k_gemm_lin(const float* __restrict__ H,      // [N][128]
                const float* __restrict__ AGG,    // [N][128]
                const float* __restrict__ W,      // [128][256] row-major
                const float* __restrict__ bias,   // [128]
                float* __restrict__ out) {        // [N][128]
  __shared__ float As[16][LDS_STRIDE];            // H tile
  __shared__ float Ag[16][LDS_STRIDE];            // AGG tile
  __shared__ float Rt[16][LDS_STRIDE];            // result tile for row-normalize
  __shared__ float rinv[16];

  const int lane = threadIdx.x & 31;
  const int wave = threadIdx.x >> 5;
  const int r0   = blockIdx.x * 16;
  const int c0   = wave * 16;
  const int mn   = lane & 15;
  const int kh   = (lane >> 4) * 2;

  stage_tile(H,   As, r0);
  stage_tile(AGG, Ag, r0);
  __syncthreads();

  const float* brow = W + (size_t)(c0 + mn) * (2 * WIDTH);

  v8f acc = {};
#pragma unroll 8
  for (int k0 = 0; k0 < WIDTH; k0 += 4) {          // K = 0..127 from H
    v2f a = *(const v2f*)&As[mn][k0 + kh];
    v2f b; b.x = brow[k0 + kh + 0]; b.y = brow[k0 + kh + 1];
    acc = __builtin_amdgcn_wmma_f32_16x16x4_f32(false, a, false, b,
                                                (short)0, acc, false, false);
  }
#pragma unroll 8
  for (int k0 = 0; k0 < WIDTH; k0 += 4) {          // K = 128..255 from AGG
    v2f a = *(const v2f*)&Ag[mn][k0 + kh];
    v2f b; b.x = brow[WIDTH + k0 + kh + 0]; b.y = brow[WIDTH + k0 + kh + 1];
    acc = __builtin_amdgcn_wmma_f32_16x16x4_f32(false, a, false, b,
                                                (short)0, acc, false, false);
  }

  const int col = c0 + mn;
  const float bb = bias[col];
#pragma unroll
  for (int v = 0; v < 8; ++v) {
    int rl = v + 8 * (lane >> 4);                  // local row 0..15
    float val = acc[v] + bb;
    Rt[rl][col] = val > 0.0f ? val : 0.0f;
  }
  __syncthreads();

  if (threadIdx.x < 16) {                          // per-row ||h||
    float s = 0.0f;
#pragma unroll 8
    for (int j = 0; j < WIDTH; ++j) { float t = Rt[threadIdx.x][j]; s += t * t; }
    rinv[threadIdx.x] = 1.0f / fmaxf(sqrtf(s), EPS);
  }
  __syncthreads();

#pragma unroll
  for (int e = 0; e < 8; ++e) {                    // 256 threads x 8 = 2048 = 16*128
    int idx = threadIdx.x * 8 + e;
    int rl  = idx >> 7;
    int cl  = idx & 127;
    out[(size_t)(r0 + rl) * WIDTH + cl] = Rt[rl][cl] * rinv[rl];
  }
}

// ---------------------------------------------------------------- final batch gather (b128)
__global__ __launch_bounds__(256)
void k_gather(const float* __restrict__ H, const int* __restrict__ node_idx,
              float* __restrict__ out) {
  int t = blockIdx.x * 256 + threadIdx.x;          // BATCH*32 threads (float4 units)
  int i = t >> 5;
  int j = t & 31;
  ((float4*)out)[t] = ((const float4*)H)[(size_t)node_idx[i] * 32 + j];
}

// ---------------------------------------------------------------- host side
extern "C" void kernel_launch(void* const* d_in, const int* in_sizes, int n_in,
                              void* d_out, int out_size, void* d_ws, size_t ws_size,
                              hipStream_t stream) {
  const int*   nbd      = (const int*)  d_in[0];
  const int*   node_idx = (const int*)  d_in[1];
  const float* feats    = (const float*)d_in[2];
  const float* W_agg    = (const float*)d_in[3];
  const float* b_agg    = (const float*)d_in[4];
  const float* W_lin    = (const float*)d_in[5];
  const float* b_lin    = (const float*)d_in[6];
  float* out = (float*)d_out;

  const size_t per = (size_t)N_NODES * WIDTH;      // 6.4M floats = 25.6 MB
  float* B0  = (float*)d_ws;                       // H ping
  float* B1  = B0 + per;                           // M / H pong
  float* AGG = B1 + per;                           // neighbor max-pool

  // H0 = zeros, scatter batch features
  const int n4 = (int)(per / 4);
  k_zero4<<<(n4 + 255) / 256, 256, 0, stream>>>((float4*)B0, n4);
  k_scatter<<<(BATCH * 32) / 256, 256, 0, stream>>>(node_idx, feats, B0);

  float* cur = B0;
  float* nxt = B1;
  for (int hop = 0; hop < DEPTH; ++hop) {
    k_gemm_agg<<<N_NODES / 16, 256, 0, stream>>>(cur, W_agg, b_agg, nxt);      // M -> nxt
    k_pool    <<<N_NODES / 8, 256, 0, stream>>>(nxt, nbd, AGG);                // AGG
    k_gemm_lin<<<N_NODES / 16, 256, 0, stream>>>(cur, AGG, W_lin, b_lin, nxt); // H' -> nxt
    float* t = cur; cur = nxt; nxt = t;
  }
  k_gather<<<(BATCH * 32) / 256, 256, 0, stream>>>(cur, node_idx, out);
}